// PatchEmbedding_85220741087419
// MI455X (gfx1250) — compile-verified
//
#include <hip/hip_runtime.h>

// MI455X (gfx1250) patch-embedding.
// Memory-bound: W (462 MB) dominates -> stream W once, stage A (patches) in LDS,
// use f32 WMMA (16x16x4) so numerics match the f32 reference exactly.

typedef float v2f __attribute__((ext_vector_type(2)));
typedef float v8f __attribute__((ext_vector_type(8)));

#define IMG    224
#define PATCHW 16
#define CH     3
#define DDIM   768
#define ROWP   14
#define NPATCH 196
#define PDIM   768     // 16*16*3
#define BATCH  32

__global__ __launch_bounds__(256)
void patch_embed_wmma_f32(const float* __restrict__ x,
                          const float* __restrict__ W,
                          const float* __restrict__ bias,
                          float* __restrict__ out)
{
    // A = patches[:, n, :]  -> 32 x 768 f32 = 96 KB LDS (WGP has 320 KB)
    __shared__ float ldsA[BATCH * PDIM];

    const int n    = blockIdx.x;       // patch index 0..195
    const int half = blockIdx.y;       // column half 0..1 (384 cols each)
    const int tid  = threadIdx.x;
    const int r    = n / ROWP;         // patch row
    const int cc   = n % ROWP;         // patch col

    // Cooperative patch extraction: A[m][p], p = c*256 + py*16 + px
    // x layout [B, C, H, W]; 16 consecutive p -> 64B contiguous in x.
    for (int idx = tid; idx < BATCH * PDIM; idx += 256) {
        const int m   = idx / PDIM;
        const int p   = idx - m * PDIM;
        const int c   = p >> 8;
        const int rem = p & 255;
        const int py  = rem >> 4;
        const int px  = rem & 15;
        ldsA[idx] = x[((size_t)(m * CH + c) * IMG + (r * PATCHW + py)) * IMG
                      + cc * PATCHW + px];
    }
    __syncthreads();

    const int wave   = tid >> 5;       // 8 waves
    const int lane   = tid & 31;
    const int hi     = lane >> 4;      // 0: lanes 0-15, 1: lanes 16-31
    const int lnlo   = lane & 15;
    const int m_base = (wave & 1) * 16;   // which 16 batch rows
    const int wp     = wave >> 1;         // 0..3: N-tile group

    const float* __restrict__ Wn   = W + (size_t)n * PDIM * DDIM;
    const float* __restrict__ Arow = &ldsA[(size_t)(m_base + lnlo) * PDIM];

    // 24 N-tiles per half, 4 wave-pairs -> 6 tiles per wave
    for (int i = 0; i < 6; ++i) {
        const int d = half * 384 + (wp + 4 * i) * 16 + lnlo;
        const float* __restrict__ bcol = Wn + d;

        v8f acc = {0.f, 0.f, 0.f, 0.f, 0.f, 0.f, 0.f, 0.f};

        // K loop: 768 in steps of 4 (f32 WMMA K=4)
        #pragma unroll 8
        for (int k0 = 0; k0 < PDIM; k0 += 4) {
            // A frag: lane<16 holds K=k0,k0+1; lane>=16 holds K=k0+2,k0+3 (ds_load_b64)
            v2f a = *(const v2f*)(Arow + k0 + 2 * hi);
            // B frag: row-striped: v0 -> K=k0+hi, v1 -> K=k0+2+hi, col = lnlo
            v2f b;
            b.x = bcol[(size_t)(k0 + hi) * DDIM];
            b.y = bcol[(size_t)(k0 + hi + 2) * DDIM];
            acc = __builtin_amdgcn_wmma_f32_16x16x4_f32(
                /*neg_a=*/false, a, /*neg_b=*/false, b,
                /*c_mod=*/(short)0, acc, /*reuse_a=*/false, /*reuse_b=*/false);
        }

        const float bv = bias[(size_t)n * DDIM + d];
        // C/D layout: VGPR j = row (m_base + j + 8*hi), col = lnlo
        #pragma unroll
        for (int j = 0; j < 8; ++j) {
            const int m = m_base + j + 8 * hi;
            out[((size_t)m * NPATCH + n) * DDIM + d] = acc[j] + bv;
        }
    }
}

extern "C" void kernel_launch(void* const* d_in, const int* in_sizes, int n_in,
                              void* d_out, int out_size, void* d_ws, size_t ws_size,
                              hipStream_t stream)
{
    const float* x  = (const float*)d_in[0];   // [32, 3, 224, 224]
    const float* W  = (const float*)d_in[1];   // [196, 768, 768]
    const float* b  = (const float*)d_in[2];   // [196, 768]
    float* out      = (float*)d_out;           // [32, 196, 768]

    dim3 grid(NPATCH, 2);
    patch_embed_wmma_f32<<<grid, 256, 0, stream>>>(x, W, b, out);
}